// RDOPT_57251914056085
// MI455X (gfx1250) — compile-verified
//
#include <hip/hip_runtime.h>
#include <math.h>

// ---------------------------------------------------------------------------
// Gauss-Newton pose refinement, CDNA5 / gfx1250 (wave32, WMMA).
//   B=32, C=3, H=W=256, 5 GN iterations.
//   Normal-equation accumulation (GtG of the N x 7 matrix [J | e]) uses
//   V_WMMA_F32_16X16X4_F32: 32 lanes each build one sample's 7-vector,
//   stage to (double-buffered) LDS, then 8 K=4 WMMAs rank-update a 16x16
//   f32 accumulator whose top-left 7x7 corner is [JtJ | Jte].
//   Param slots 7..15 of every staging row are zeroed ONCE at kernel entry
//   and never rewritten, so the WMMA operands come straight out of ds_load
//   with no per-chunk masking.
//   Working set (f_inp+depth+face+base ~ 43 MB) is L2-resident (192 MB),
//   so the 5 GN passes cost ~one HBM sweep total.
// ---------------------------------------------------------------------------

#define BATCH 32
#define CCH 3
#define HH 256
#define WW 256
#define HWPIX (HH * WW)            // 65536
#define NSAMP (CCH * HWPIX)        // 196608 pixel-channel samples per batch
#define NGN 5
#define THREADS 256
#define WAVES (THREADS / 32)
#define BLOCKS_PER_B 96
#define PASSES (NSAMP / (BLOCKS_PER_B * THREADS))  // = 8, exact

typedef __attribute__((ext_vector_type(2))) float v2f;
typedef __attribute__((ext_vector_type(8))) float v8f;

// ---------------------------------------------------------------------------
__global__ void gn_init(const float* __restrict__ x_ini, float* __restrict__ x_cur) {
    int i = blockIdx.x * blockDim.x + threadIdx.x;
    if (i < BATCH * 6) x_cur[i] = x_ini[i];
}

__global__ void gn_zero(float* __restrict__ acc) {
    int i = blockIdx.x * blockDim.x + threadIdx.x;
    if (i < BATCH * 64) acc[i] = 0.0f;
}

__global__ void gn_copy(const float* __restrict__ x_cur, float* __restrict__ out) {
    int i = blockIdx.x * blockDim.x + threadIdx.x;
    if (i < BATCH * 6) out[i] = x_cur[i];
}

// ---------------------------------------------------------------------------
// Accumulate per-batch 7x7 normal matrix:  acc[b][m][n] = sum_k g_k[m]*g_k[n]
// with g = [J_row(6), e].  J is masked by face validity (reference masks Jc).
// ---------------------------------------------------------------------------
__global__ __launch_bounds__(THREADS)
void gn_accum(const float* __restrict__ f_inp,   // (B,C,H,W)
              const float* __restrict__ depth,   // (B,H,W)
              const int*   __restrict__ face,    // (B,H,W)
              const float* __restrict__ Kmat,    // (B,3,3)
              const float* __restrict__ bbox,    // (B,2)
              const float* __restrict__ Wp,      // (3,6)
              const float* __restrict__ base,    // (C,H,W)
              const float* __restrict__ x_cur,   // (B,6)
              float*       __restrict__ acc_out) // (B,64)
{
    // double-buffered per-wave staging: [buf][wave][sample][param(17-stride)]
    __shared__ float lds[2 * WAVES * 32 * 17];

    const int b    = blockIdx.y;
    const int blk  = blockIdx.x;
    const int tid  = threadIdx.x;
    const int lane = tid & 31;
    const int wave = tid >> 5;

    // Zero the pad region (param slots 7..16) of both buffers once; the
    // per-pass stores only touch slots 0..6, so rows 7..15 of the WMMA A
    // operand stay zero for the whole kernel -> no per-chunk masking.
    {
        float* r0 = &lds[(0 * WAVES * 32 + wave * 32 + lane) * 17];
        float* r1 = &lds[(1 * WAVES * 32 + wave * 32 + lane) * 17];
#pragma unroll
        for (int j = 7; j < 17; ++j) { r0[j] = 0.0f; r1[j] = 0.0f; }
    }

    // -------- uniform per-batch pose constants --------
    const float x0 = x_cur[b * 6 + 0], x1 = x_cur[b * 6 + 1], x2 = x_cur[b * 6 + 2];
    const float t0 = x_cur[b * 6 + 3], t1 = x_cur[b * 6 + 4], t2 = x_cur[b * 6 + 5];
    float th[3];
#pragma unroll
    for (int c = 0; c < 3; ++c) {
        th[c] = tanhf(x0 * Wp[c * 6 + 0] + x1 * Wp[c * 6 + 1] + x2 * Wp[c * 6 + 2] +
                      t0 * Wp[c * 6 + 3] + t1 * Wp[c * 6 + 4] + t2 * Wp[c * 6 + 5]);
    }
    const float fx = Kmat[b * 9 + 0], cx = Kmat[b * 9 + 2];
    const float fy = Kmat[b * 9 + 4], cy = Kmat[b * 9 + 5];
    const float rfx = 1.0f / fx, rfy = 1.0f / fy;
    const float bx = bbox[b * 2 + 0], by = bbox[b * 2 + 1];

    const float* fb = f_inp + (size_t)b * NSAMP;
    const float* db = depth + (size_t)b * HWPIX;
    const int*   mb = face + (size_t)b * HWPIX;

    v8f acc = {};                       // 16x16 f32 WMMA accumulator
    const int m    = lane & 15;         // param index this lane feeds
    const int half = lane >> 4;         // K-pair selector per the 16x4 A layout

    for (int pass = 0; pass < PASSES; ++pass) {
        // channel-major sample id: s = c*HW + p  -> consecutive lanes hit
        // consecutive pixels of one channel plane (fully coalesced).
        const int s = (blk * PASSES + pass) * THREADS + tid;
        const int c = s >> 16;
        const int p = s & (HWPIX - 1);
        const int h = p >> 8;
        const int w = p & 255;

        // WGP-scope streaming prefetch (global_prefetch_b8, near-cache)
        __builtin_prefetch(fb + s + THREADS, 0, 3);
        __builtin_prefetch(db + p + THREADS, 0, 3);

        const float z     = db[p];
        const float maskf = (mb[p] >= 0) ? 1.0f : 0.0f;
        const float inv_z = 1.0f / z;

        const float u  = bx + (float)w;
        const float v  = by + (float)h;
        const float xc = (u - cx) * z * rfx;
        const float yc = (v - cy) * z * rfy;

        // Jp rows (projection Jacobian wrt camera point); note
        // fx*xc*inv_z^2 == (u-cx)*inv_z, so a2/b2 simplify.
        const float a0 = fx * inv_z;
        const float a2 = -(u - cx) * inv_z;
        const float b1 = fy * inv_z;
        const float b2 = -(v - cy) * inv_z;

        const float px = xc - t0, py = yc - t1, pz = z - t2;

        // J_rot = -Jp * skew(P)
        float Jc0[6], Jc1[6];
        Jc0[0] = a2 * py;             Jc1[0] = -b1 * pz + b2 * py;
        Jc0[1] = a0 * pz - a2 * px;   Jc1[1] = -b2 * px;
        Jc0[2] = -a0 * py;            Jc1[2] = b1 * px;
        Jc0[3] = a0;                  Jc1[3] = 0.0f;
        Jc0[4] = 0.0f;                Jc1[4] = b1;
        Jc0[5] = a2;                  Jc1[5] = b2;

        // spatial gradient of base_feat (f_rend differs by a constant, so
        // grad(f_rend) == grad(base)).  Branch-free jnp.gradient edge rule:
        // clamp neighbor indices, scale 0.5 interior / 1.0 at edges.
        const float* brow = base + c * HWPIX + h * WW;
        const int   wm = (w > 0)      ? w - 1 : 0;
        const int   wp = (w < WW - 1) ? w + 1 : WW - 1;
        const float sx = (wp - wm == 2) ? 0.5f : 1.0f;
        const float gx = (brow[wp] - brow[wm]) * sx;

        const float* bcol = base + c * HWPIX + w;
        const int   hm = (h > 0)      ? h - 1 : 0;
        const int   hp = (h < HH - 1) ? h + 1 : HH - 1;
        const float sy = (hp - hm == 2) ? 0.5f : 1.0f;
        const float gy = (bcol[hp * WW] - bcol[hm * WW]) * sy;

        const float e = fb[s] - brow[w] - th[c];

        // g = [ (gx*Jc0 + gy*Jc1) * mask , e ]   -> staged to this pass's buf
        const int    buf = pass & 1;
        float* myl = &lds[(buf * WAVES * 32 + wave * 32 + lane) * 17];
#pragma unroll
        for (int j = 0; j < 6; ++j) myl[j] = (gx * Jc0[j] + gy * Jc1[j]) * maskf;
        myl[6] = e;

        __syncthreads();   // single barrier per pass (double-buffered LDS)

        // 8 rank-4 updates: D += A * A (A == B by symmetry of GtG).
        const float* wl = &lds[(buf * WAVES * 32 + wave * 32) * 17];
#pragma unroll
        for (int ch = 0; ch < 8; ++ch) {
            const int k0 = ch * 4 + 2 * half;
            v2f A = {wl[k0 * 17 + m], wl[(k0 + 1) * 17 + m]};
            acc = __builtin_amdgcn_wmma_f32_16x16x4_f32(
                      false, A, false, A, (short)0, acc, false, false);
        }
    }

    // D layout (f32 16x16): lane L<16 holds N=L, VGPR r holds M=r.
    // Needed corner: M,N in [0,7).
    if (lane < 7) {
        float* ab = acc_out + b * 64;
#pragma unroll
        for (int r = 0; r < 7; ++r) atomicAdd(&ab[r * 8 + lane], acc[r]);
    }
}

// ---------------------------------------------------------------------------
// One thread per batch: solve (JtJ + damping*I) dx = Jte, update x.
// ---------------------------------------------------------------------------
__global__ void gn_solve(const float* __restrict__ acc,
                         const float* __restrict__ lam,
                         float* __restrict__ x_cur)
{
    const int b = threadIdx.x;
    if (b >= BATCH) return;

    const float damping = log1pf(expf(lam[0])) + 1e-6f;
    const float* ab = acc + b * 64;

    float A[6][7];
#pragma unroll
    for (int i = 0; i < 6; ++i) {
#pragma unroll
        for (int j = 0; j < 6; ++j)
            A[i][j] = ab[i * 8 + j] + ((i == j) ? damping : 0.0f);
        A[i][6] = ab[i * 8 + 6];       // Jte column of GtG
    }

    // Gaussian elimination with partial pivoting
    for (int k = 0; k < 6; ++k) {
        int piv = k;
        float best = fabsf(A[k][k]);
        for (int r = k + 1; r < 6; ++r) {
            float vv = fabsf(A[r][k]);
            if (vv > best) { best = vv; piv = r; }
        }
        if (piv != k) {
            for (int j = k; j < 7; ++j) {
                float t = A[k][j]; A[k][j] = A[piv][j]; A[piv][j] = t;
            }
        }
        const float inv = 1.0f / A[k][k];
        for (int r = k + 1; r < 6; ++r) {
            const float f = A[r][k] * inv;
            for (int j = k; j < 7; ++j) A[r][j] -= f * A[k][j];
        }
    }
    float dx[6];
    for (int i = 5; i >= 0; --i) {
        float s = A[i][6];
        for (int j = i + 1; j < 6; ++j) s -= A[i][j] * dx[j];
        dx[i] = s / A[i][i];
    }
#pragma unroll
    for (int i = 0; i < 6; ++i) x_cur[b * 6 + i] += dx[i];
}

// ---------------------------------------------------------------------------
extern "C" void kernel_launch(void* const* d_in, const int* in_sizes, int n_in,
                              void* d_out, int out_size, void* d_ws, size_t ws_size,
                              hipStream_t stream) {
    const float* f_inp = (const float*)d_in[0];
    const float* x_ini = (const float*)d_in[1];
    const float* depth = (const float*)d_in[2];
    const int*   face  = (const int*)d_in[3];
    const float* Kmat  = (const float*)d_in[4];
    const float* bbox  = (const float*)d_in[5];
    const float* Wp    = (const float*)d_in[6];
    const float* base  = (const float*)d_in[7];
    const float* lam   = (const float*)d_in[8];

    float* x_cur = (float*)d_ws;          // 192 floats (pad to 256)
    float* acc   = x_cur + 256;           // B*64 floats

    gn_init<<<1, 256, 0, stream>>>(x_ini, x_cur);

    for (int it = 0; it < NGN; ++it) {
        gn_zero<<<(BATCH * 64 + 255) / 256, 256, 0, stream>>>(acc);
        dim3 grid(BLOCKS_PER_B, BATCH);
        gn_accum<<<grid, THREADS, 0, stream>>>(f_inp, depth, face, Kmat, bbox,
                                               Wp, base, x_cur, acc);
        gn_solve<<<1, 32, 0, stream>>>(acc, lam, x_cur);
    }
    gn_copy<<<1, 256, 0, stream>>>(x_cur, (float*)d_out);
}